// ArmaNet02_60687887893295
// MI455X (gfx1250) — compile-verified
//
#include <hip/hip_runtime.h>
#include <math.h>

#define NN 100000
#define NE 1600000
#define KK 3
#define HH 16
#define F1 48              // KK*HH
#define TILES (NN / 16)    // 6250 (N divisible by 16)
#define UNITS (TILES * KK) // 18750 wave-units for the WMMA matmul

typedef float v2f __attribute__((ext_vector_type(2)));
typedef float v8f __attribute__((ext_vector_type(8)));

// Native CDNA5 no-return fp32 atomic add at device scope (RMW performed in L2).
// Guarantees we never fall back to a CAS loop: fire-and-forget, STOREcnt-tracked.
__device__ __forceinline__ void gAddF32(float* p, float v) {
  asm volatile("global_atomic_add_f32 %0, %1, off scope:SCOPE_DEV"
               :
               : "v"(p), "v"(v)
               : "memory");
}

// ---------------- utility kernels ----------------
__global__ void k_zero(float* p, int n) {
  int i = blockIdx.x * blockDim.x + threadIdx.x;
  if (i < n) p[i] = 0.f;
}
__global__ void k_copy(float* __restrict__ dst, const float* __restrict__ src, int n) {
  int i = blockIdx.x * blockDim.x + threadIdx.x;
  if (i < n) dst[i] = src[i];
}

// ---------------- graph norm ----------------
// deg accumulation + int64 -> int32 edge index conversion (halves index bytes)
__global__ void k_deg(const long long* __restrict__ ei, const float* __restrict__ ew,
                      float* __restrict__ deg, int* __restrict__ s32, int* __restrict__ d32) {
  int e = blockIdx.x * blockDim.x + threadIdx.x;
  if (e >= NE) return;
  int s = (int)ei[e];
  int d = (int)ei[NE + e];
  s32[e] = s;
  d32[e] = d;
  gAddF32(&deg[d], ew[e]);
}

__global__ void k_dis(float* deg) {
  int n = blockIdx.x * blockDim.x + threadIdx.x;
  if (n >= NN) return;
  float dg = deg[n];
  deg[n] = (dg > 0.f) ? (1.f / sqrtf(dg)) : 0.f;  // in-place: deg -> dis
}

__global__ void k_norm(const int* __restrict__ s32, const int* __restrict__ d32,
                       const float* __restrict__ ew, const float* __restrict__ dis,
                       float* __restrict__ norm) {
  int e = blockIdx.x * blockDim.x + threadIdx.x;
  if (e >= NE) return;
  norm[e] = dis[s32[e]] * ew[e] * dis[d32[e]];
}

// ---------------- conv1 (1 -> 16, K stacks) ----------------
// state layout: [n * 48 + k * 16 + h], f32

__global__ void k_init1(const float* __restrict__ x, const float* __restrict__ w1i,
                        float* __restrict__ M) {
  int i = blockIdx.x * blockDim.x + threadIdx.x;
  if (i >= NN * F1) return;
  int n = i / F1, f = i - n * F1;
  M[i] = x[n] * w1i[f];  // einsum('n1,k1h->nkh')
}

// prefill accumulation buffer with root = x*w1_root + b  (so scatter just adds)
__global__ void k_prefill1(const float* __restrict__ x, const float* __restrict__ w1r,
                           const float* __restrict__ w1b, float* __restrict__ A) {
  int i = blockIdx.x * blockDim.x + threadIdx.x;
  if (i >= NN * F1) return;
  int n = i / F1, f = i - n * F1;
  A[i] = x[n] * w1r[f] + w1b[f];
}

// edge gather/scatter: A[dst] += M[src] * norm ; one thread = one float4 chunk
__global__ void k_scatter1(const float* __restrict__ M, const int* __restrict__ s32,
                           const int* __restrict__ d32, const float* __restrict__ norm,
                           float* __restrict__ A) {
  int i = blockIdx.x * blockDim.x + threadIdx.x;
  if (i >= NE * 12) return;  // 12 float4 chunks of 48 floats
  int e = i / 12;
  int q = i - e * 12;
  int s = s32[e], d = d32[e];
  float nm = norm[e];
  float4 v = *(const float4*)(M + (size_t)s * F1 + q * 4);
  float* ap = A + (size_t)d * F1 + q * 4;
  gAddF32(ap + 0, v.x * nm);
  gAddF32(ap + 1, v.y * nm);
  gAddF32(ap + 2, v.z * nm);
  gAddF32(ap + 3, v.w * nm);
}

// per-stack dense 16x16 matmul via chained V_WMMA_F32_16X16X4_F32
// one wave computes a 16-node x 16-channel tile for one stack k.
// relu of the previous iteration is fused into the A-matrix load.
__global__ void __launch_bounds__(256) k_mm1(const float* __restrict__ Ain,
                                             const float* __restrict__ w1w,
                                             float* __restrict__ M) {
  int wave = threadIdx.x >> 5;
  int lane = threadIdx.x & 31;
  int unit = blockIdx.x * 8 + wave;   // wave-uniform -> EXEC all-ones past guard
  if (unit >= UNITS) return;
  int tile = unit / KK;
  int k = unit - tile * KK;
  int node0 = tile * 16;
  int half = lane >> 4;   // lanes 16-31 supply K=2,3 of each 4-wide K slab
  int r = lane & 15;      // A row (node) / B column (out channel)
  const float* wk = w1w + k * 256;                               // W_k[i][o], 16x16
  const float* arow = Ain + (size_t)(node0 + r) * F1 + k * HH;   // node row, stack k
  v8f c = {};
#pragma unroll
  for (int j = 0; j < 4; ++j) {
    int kofs = 4 * j + 2 * half;
    v2f a, b;
    a.x = fmaxf(arow[kofs], 0.f);        // ISA A layout: VGPR0=K0(/K2), VGPR1=K1(/K3)
    a.y = fmaxf(arow[kofs + 1], 0.f);
    b.x = wk[kofs * 16 + r];             // B row K, col r
    b.y = wk[(kofs + 1) * 16 + r];
    c = __builtin_amdgcn_wmma_f32_16x16x4_f32(false, a, false, b, (short)0, c,
                                              false, false);
  }
  // D layout: VGPR vr = (node0 + vr + 8*half), channel = lane&15
#pragma unroll
  for (int vr = 0; vr < 8; ++vr) {
    int nd = node0 + vr + 8 * half;
    M[(size_t)nd * F1 + k * HH + r] = c[vr];
  }
}

// relu + mean over stacks -> h1[N,16]; accumulate BN sum/sumsq via LDS+global atomics
__global__ void k_finalize1(const float* __restrict__ A, float* __restrict__ h1,
                            float* __restrict__ bn) {
  __shared__ float ssum[16];
  __shared__ float ssq[16];
  int t = threadIdx.x;
  if (t < 16) { ssum[t] = 0.f; ssq[t] = 0.f; }
  __syncthreads();
  int n = blockIdx.x * blockDim.x + t;
  if (n < NN) {
    const float* an = A + (size_t)n * F1;
#pragma unroll
    for (int c = 0; c < 16; ++c) {
      float s = fmaxf(an[c], 0.f) + fmaxf(an[16 + c], 0.f) + fmaxf(an[32 + c], 0.f);
      s *= (1.f / 3.f);
      h1[(size_t)n * 16 + c] = s;
      atomicAdd(&ssum[c], s);      // ds_add_f32
      atomicAdd(&ssq[c], s * s);
    }
  }
  __syncthreads();
  if (t < 16) {
    gAddF32(&bn[t], ssum[t]);
    gAddF32(&bn[16 + t], ssq[t]);
  }
}

// derive BN affine: a = g/sqrt(var+eps), b = beta - mu*a  (biased var, matches jnp.var)
__global__ void k_bnparam(float* bn, const float* __restrict__ g,
                          const float* __restrict__ b) {
  int c = threadIdx.x;
  if (c >= 16) return;
  float mu = bn[c] * (1.f / NN);
  float var = bn[16 + c] * (1.f / NN) - mu * mu;
  float a = g[c] / sqrtf(var + 1e-5f);
  bn[32 + c] = a;
  bn[48 + c] = b[c] - mu * a;
}

// ---------------- conv2 (16 -> 1, K stacks), state [n*3 + k] ----------------
__global__ void k_conv2init(const float* __restrict__ h1, const float* __restrict__ bn,
                            const float* __restrict__ w2i, const float* __restrict__ w2r,
                            const float* __restrict__ w2b, float* __restrict__ m2,
                            float* __restrict__ r2) {
  int n = blockIdx.x * blockDim.x + threadIdx.x;
  if (n >= NN) return;
  float hb[16];
#pragma unroll
  for (int c = 0; c < 16; ++c)
    hb[c] = fmaxf(bn[32 + c] * h1[(size_t)n * 16 + c] + bn[48 + c], 0.f);
#pragma unroll
  for (int k = 0; k < KK; ++k) {
    float si = 0.f, sr = 0.f;
#pragma unroll
    for (int c = 0; c < 16; ++c) {
      si += hb[c] * w2i[k * 16 + c];
      sr += hb[c] * w2r[k * 16 + c];
    }
    m2[n * KK + k] = si;
    r2[n * KK + k] = sr + w2b[k];
  }
}

// edge scatter for conv2; the 1x1 per-stack weight (t>0) is folded into the gather
__global__ void k_scatter2(const float* __restrict__ prev, const int* __restrict__ s32,
                           const int* __restrict__ d32, const float* __restrict__ norm,
                           const float* __restrict__ w2w, int mulw,
                           float* __restrict__ acc) {
  int e = blockIdx.x * blockDim.x + threadIdx.x;
  if (e >= NE) return;
  int s = s32[e], d = d32[e];
  float nm = norm[e];
#pragma unroll
  for (int k = 0; k < KK; ++k) {
    float v = prev[s * KK + k];
    if (mulw) v *= w2w[k];
    gAddF32(&acc[d * KK + k], v * nm);
  }
}

__global__ void k_final(const float* __restrict__ acc, const float* __restrict__ lw,
                        const float* __restrict__ lb, float* __restrict__ out) {
  int n = blockIdx.x * blockDim.x + threadIdx.x;
  if (n >= NN) return;
  float s = (acc[n * 3] + acc[n * 3 + 1] + acc[n * 3 + 2]) * (1.f / 3.f);
  float y = s * lw[0] + lb[0];
  out[n] = 1.f / (1.f + expf(-y));
}

// ---------------- host launcher ----------------
extern "C" void kernel_launch(void* const* d_in, const int* in_sizes, int n_in,
                              void* d_out, int out_size, void* d_ws, size_t ws_size,
                              hipStream_t stream) {
  const float* x = (const float*)d_in[0];
  const long long* ei = (const long long*)d_in[1];  // int64 edge_index
  const float* ew = (const float*)d_in[2];
  const float* w1i = (const float*)d_in[3];
  const float* w1w = (const float*)d_in[4];
  const float* w1r = (const float*)d_in[5];
  const float* w1b = (const float*)d_in[6];
  const float* bng = (const float*)d_in[7];
  const float* bnb = (const float*)d_in[8];
  const float* w2i = (const float*)d_in[9];
  const float* w2w = (const float*)d_in[10];
  const float* w2r = (const float*)d_in[11];
  const float* w2b = (const float*)d_in[12];
  const float* lw = (const float*)d_in[13];
  const float* lb = (const float*)d_in[14];
  float* out = (float*)d_out;

  // workspace carve-up (256B aligned), total ~88 MB — L2-resident on MI455X
  char* ws = (char*)d_ws;
  auto aln = [](size_t v) { return (v + 255) & ~(size_t)255; };
  size_t off = 0;
  float* norm = (float*)(ws + off); off = aln(off + (size_t)NE * 4);
  int* s32 = (int*)(ws + off);      off = aln(off + (size_t)NE * 4);
  int* d32 = (int*)(ws + off);      off = aln(off + (size_t)NE * 4);
  float* M1 = (float*)(ws + off);   off = aln(off + (size_t)NN * F1 * 4);
  float* A1a = (float*)(ws + off);  off = aln(off + (size_t)NN * F1 * 4);
  float* A1b = (float*)(ws + off);  off = aln(off + (size_t)NN * F1 * 4);
  float* h1 = (float*)(ws + off);   off = aln(off + (size_t)NN * 16 * 4);
  float* r2 = (float*)(ws + off);   off = aln(off + (size_t)NN * 3 * 4);
  float* m2 = (float*)(ws + off);   off = aln(off + (size_t)NN * 3 * 4);
  float* a2a = (float*)(ws + off);  off = aln(off + (size_t)NN * 3 * 4);
  float* a2b = (float*)(ws + off);  off = aln(off + (size_t)NN * 3 * 4);
  float* deg = (float*)(ws + off);  off = aln(off + (size_t)NN * 4);
  float* bn = (float*)(ws + off);   off = aln(off + 64 * 4);

  const int B = 256;
  auto gr = [](long n) { return (int)((n + 255) / 256); };

  // graph norm
  k_zero<<<gr(NN), B, 0, stream>>>(deg, NN);
  k_zero<<<1, 64, 0, stream>>>(bn, 64);
  k_deg<<<gr(NE), B, 0, stream>>>(ei, ew, deg, s32, d32);
  k_dis<<<gr(NN), B, 0, stream>>>(deg);
  k_norm<<<gr(NE), B, 0, stream>>>(s32, d32, ew, deg, norm);

  // conv1: t = 0..3, ping-pong A1a/A1b, relu fused into next consumer
  k_init1<<<gr((long)NN * F1), B, 0, stream>>>(x, w1i, M1);
  k_prefill1<<<gr((long)NN * F1), B, 0, stream>>>(x, w1r, w1b, A1a);
  k_scatter1<<<gr((long)NE * 12), B, 0, stream>>>(M1, s32, d32, norm, A1a);

  k_mm1<<<(UNITS + 7) / 8, 256, 0, stream>>>(A1a, w1w, M1);
  k_prefill1<<<gr((long)NN * F1), B, 0, stream>>>(x, w1r, w1b, A1b);
  k_scatter1<<<gr((long)NE * 12), B, 0, stream>>>(M1, s32, d32, norm, A1b);

  k_mm1<<<(UNITS + 7) / 8, 256, 0, stream>>>(A1b, w1w, M1);
  k_prefill1<<<gr((long)NN * F1), B, 0, stream>>>(x, w1r, w1b, A1a);
  k_scatter1<<<gr((long)NE * 12), B, 0, stream>>>(M1, s32, d32, norm, A1a);

  k_mm1<<<(UNITS + 7) / 8, 256, 0, stream>>>(A1a, w1w, M1);
  k_prefill1<<<gr((long)NN * F1), B, 0, stream>>>(x, w1r, w1b, A1b);
  k_scatter1<<<gr((long)NE * 12), B, 0, stream>>>(M1, s32, d32, norm, A1b);

  // BN stats + affine, then conv2 init
  k_finalize1<<<gr(NN), B, 0, stream>>>(A1b, h1, bn);
  k_bnparam<<<1, 32, 0, stream>>>(bn, bng, bnb);
  k_conv2init<<<gr(NN), B, 0, stream>>>(h1, bn, w2i, w2r, w2b, m2, r2);

  // conv2: t = 0..3
  k_copy<<<gr((long)NN * 3), B, 0, stream>>>(a2a, r2, NN * 3);
  k_scatter2<<<gr(NE), B, 0, stream>>>(m2, s32, d32, norm, w2w, 0, a2a);
  k_copy<<<gr((long)NN * 3), B, 0, stream>>>(a2b, r2, NN * 3);
  k_scatter2<<<gr(NE), B, 0, stream>>>(a2a, s32, d32, norm, w2w, 1, a2b);
  k_copy<<<gr((long)NN * 3), B, 0, stream>>>(a2a, r2, NN * 3);
  k_scatter2<<<gr(NE), B, 0, stream>>>(a2b, s32, d32, norm, w2w, 1, a2a);
  k_copy<<<gr((long)NN * 3), B, 0, stream>>>(a2b, r2, NN * 3);
  k_scatter2<<<gr(NE), B, 0, stream>>>(a2a, s32, d32, norm, w2w, 1, a2b);

  k_final<<<gr(NN), B, 0, stream>>>(a2b, lw, lb, out);
}